// TrainableSegInvPositionalEncoding_11416023072873
// MI455X (gfx1250) — compile-verified
//
#include <hip/hip_runtime.h>

// ---------------------------------------------------------------------------
// TrainableSegInvPositionalEncoding for MI455X (gfx1250) — 3-kernel pipeline.
//
// Footprint (~125MB) fits in the 192MB L2; bound by L2 atomic throughput.
//   ws layout: float2[T]  where .x = byte_sum, .y = byte_cnt (as f32, exact)
//   1) zero    : ws (2*T words, T read from device scalar) + out (N words)
//   2) pass1   : per edge: v = params[pidx]; ws[bpos] += {v, 1.0f}  (2 atomics,
//                same 64B line)
//   3) pass2   : per edge: {s,c} = ws[bpos] (one B64 gather);
//                out[tidx] += s/c.   Safe: every gathered bpos was hit in
//                pass1 by this very edge, so c >= 1 — the reference's
//                cnt==0 branch is unreachable on the gather path.
//
// occ_param_idx / occ_token_idx are single-use streams -> nontemporal (TH=NT)
// B128 loads keep L2 for ws + occ_byte_pos (which is read in both passes).
// ---------------------------------------------------------------------------

typedef int v4i __attribute__((ext_vector_type(4)));  // native vector: NT-loadable

__device__ __forceinline__ void atom_add_f32(float* p, float v) {
  // relaxed + agent scope -> native global_atomic_add_f32 (no-return, STOREcnt)
  __hip_atomic_fetch_add(p, v, __ATOMIC_RELAXED, __HIP_MEMORY_SCOPE_AGENT);
}

// Zero ws ([sum,cnt] pairs: 2*T words, T from device scalar) and out (N words).
__global__ void zero_kernel(unsigned* __restrict__ ws_words,
                            float* __restrict__ out, int n,
                            const int* __restrict__ d_text_len) {
  const unsigned t      = (unsigned)*d_text_len;
  const unsigned total  = 2u * t;
  const unsigned n4     = total >> 2;
  const unsigned tid    = blockIdx.x * blockDim.x + threadIdx.x;
  const unsigned stride = gridDim.x * blockDim.x;

  uint4* __restrict__ w4 = (uint4*)ws_words;
  for (unsigned k = tid; k < n4; k += stride)
    w4[k] = make_uint4(0u, 0u, 0u, 0u);
  for (unsigned k = (n4 << 2) + tid; k < total; k += stride)  // tail (<4 words)
    ws_words[k] = 0u;
  for (unsigned k = tid; k < (unsigned)n; k += stride)
    out[k] = 0.0f;
}

// Pass 1: gather params, scatter-add {sum, 1.0} into interleaved float2 ws.
// 4 edges per lane via B128 loads; single-use pidx stream uses NT hint.
__global__ void scatter_bytes_kernel(const float* __restrict__ params,
                                     const int*   __restrict__ occ_param_idx,
                                     const int*   __restrict__ occ_byte_pos,
                                     float*       __restrict__ ws,  // float2[T]
                                     int e) {
  const int e4     = e >> 2;
  const int tid    = blockIdx.x * blockDim.x + threadIdx.x;
  const int stride = gridDim.x * blockDim.x;
  const v4i* __restrict__ pi4 = (const v4i*)occ_param_idx;
  const v4i* __restrict__ bp4 = (const v4i*)occ_byte_pos;

  for (int k = tid; k < e4; k += stride) {
    const v4i pi = __builtin_nontemporal_load(&pi4[k]);   // read-once stream
    const v4i bp = bp4[k];                                // re-read in pass 2
    const float vx = params[pi.x];
    const float vy = params[pi.y];
    const float vz = params[pi.z];
    const float vw = params[pi.w];
    float* ax = ws + 2u * (unsigned)bp.x;
    float* ay = ws + 2u * (unsigned)bp.y;
    float* az = ws + 2u * (unsigned)bp.z;
    float* aw = ws + 2u * (unsigned)bp.w;
    atom_add_f32(ax, vx);  atom_add_f32(ax + 1, 1.0f);
    atom_add_f32(ay, vy);  atom_add_f32(ay + 1, 1.0f);
    atom_add_f32(az, vz);  atom_add_f32(az + 1, 1.0f);
    atom_add_f32(aw, vw);  atom_add_f32(aw + 1, 1.0f);
  }
  for (int k = (e4 << 2) + tid; k < e; k += stride) {     // tail (<4 edges)
    const int p = occ_param_idx[k];
    float* a = ws + 2u * (unsigned)occ_byte_pos[k];
    atom_add_f32(a, params[p]);
    atom_add_f32(a + 1, 1.0f);
  }
}

// Pass 2 (mean fused): one B64 gather of {sum,cnt}, divide, scatter-add to out.
__global__ void token_pos_kernel(const float2* __restrict__ ws,  // float2[T]
                                 const int*    __restrict__ occ_byte_pos,
                                 const int*    __restrict__ occ_token_idx,
                                 float*        __restrict__ out,
                                 int e) {
  const int e4     = e >> 2;
  const int tid    = blockIdx.x * blockDim.x + threadIdx.x;
  const int stride = gridDim.x * blockDim.x;
  const v4i* __restrict__ bp4 = (const v4i*)occ_byte_pos;
  const v4i* __restrict__ ti4 = (const v4i*)occ_token_idx;

  for (int k = tid; k < e4; k += stride) {
    const v4i bp = bp4[k];
    const v4i ti = __builtin_nontemporal_load(&ti4[k]);   // read-once stream
    const float2 sx = ws[bp.x];
    const float2 sy = ws[bp.y];
    const float2 sz = ws[bp.z];
    const float2 sw = ws[bp.w];
    // cnt >= 1 guaranteed for every gathered position (this edge hit it in pass 1)
    atom_add_f32(&out[ti.x], sx.x / sx.y);
    atom_add_f32(&out[ti.y], sy.x / sy.y);
    atom_add_f32(&out[ti.z], sz.x / sz.y);
    atom_add_f32(&out[ti.w], sw.x / sw.y);
  }
  for (int k = (e4 << 2) + tid; k < e; k += stride) {     // tail
    const float2 s = ws[occ_byte_pos[k]];
    atom_add_f32(&out[occ_token_idx[k]], s.x / s.y);
  }
}

extern "C" void kernel_launch(void* const* d_in, const int* in_sizes, int n_in,
                              void* d_out, int out_size, void* d_ws, size_t ws_size,
                              hipStream_t stream) {
  const float* flat_params   = (const float*)d_in[0];
  const int*   occ_param_idx = (const int*)d_in[1];
  const int*   occ_byte_pos  = (const int*)d_in[2];
  const int*   occ_token_idx = (const int*)d_in[3];
  const int*   d_text_len    = (const int*)d_in[4];   // device scalar (T)
  // d_in[5] = num_tokens device scalar; host-side N == out_size.

  const int E = in_sizes[1];
  const int N = out_size;

  float* out = (float*)d_out;
  float* ws  = (float*)d_ws;   // float2[T]: interleaved {byte_sum, byte_cnt}

  const int threads = 256;                               // 8 wave32 per block
  const int gridT   = 8192;                              // grid-stride, T device-only
  const int gridE   = (E / 4 + threads - 1) / threads;   // 1 lane per 4 edges

  // Stream order serializes the 3 pipeline stages (graph-capture safe).
  zero_kernel<<<gridT, threads, 0, stream>>>((unsigned*)ws, out, N, d_text_len);
  scatter_bytes_kernel<<<gridE, threads, 0, stream>>>(flat_params, occ_param_idx,
                                                      occ_byte_pos, ws, E);
  token_pos_kernel<<<gridE, threads, 0, stream>>>((const float2*)ws, occ_byte_pos,
                                                  occ_token_idx, out, E);
}